// StandaloneDetrTransformerDecoderLayer_2860448219638
// MI455X (gfx1250) — compile-verified
//
#include <hip/hip_runtime.h>
#include <math.h>

// Problem constants (from reference)
#define BS   8
#define NQ   1000
#define NV   20000
#define EE   256
#define HH   8
#define DH   32
#define PP   4
#define FFND 512
#define SH   100
#define SW   200
#define QT   63          // ceil(NQ/16)

typedef float v2f __attribute__((ext_vector_type(2)));
typedef float v8f __attribute__((ext_vector_type(8)));

// ---------------------------------------------------------------------------
// f32 WMMA 16x16x4:  D(16x16,f32) = A(16x4,f32) * B(4x16,f32) + C
// A frag: lane l holds A[m=l%16][k = 2*(l/16)+v], v=0..1  (contiguous pair)
// B frag: lane l holds B[k = 2*(l/16)+v][n=l%16]
// C frag: lane l holds C[m = v+8*(l/16)][n=l%16], v=0..7
// ---------------------------------------------------------------------------
__device__ __forceinline__ v8f wmma4(v2f a, v2f b, v8f c) {
  return __builtin_amdgcn_wmma_f32_16x16x4_f32(false, a, false, b, (short)0, c,
                                               false, false);
}

__device__ __forceinline__ float half_max16(float x) {
  #pragma unroll
  for (int m = 1; m < 16; m <<= 1) x = fmaxf(x, __shfl_xor(x, m, 32));
  return x;
}
__device__ __forceinline__ float half_sum16(float x) {
  #pragma unroll
  for (int m = 1; m < 16; m <<= 1) x += __shfl_xor(x, m, 32);
  return x;
}
__device__ __forceinline__ float wave_sum32(float x) {
  #pragma unroll
  for (int m = 1; m < 32; m <<= 1) x += __shfl_xor(x, m, 32);
  return x;
}

// ---------------------------------------------------------------------------
// WMMA GEMM, 32x64 macro-tile per wave (2 M-tiles x 4 N-tiles, 8 accums):
//   C[M x N] = A[M x K] * W[N x K]^T + bias, optional ReLU.
// Double-buffered: the 6 fragment loads for step k+4 are issued before the
// 8 WMMAs of step k, so the loads retire under a full WMMA shadow instead of
// gating each WMMA pair with s_wait_loadcnt 0.
// M multiple of 32, N multiple of 16, K multiple of 4.
// block = (32,4); grid = (M/32, ceil(ceil(N/64)/4))
// ---------------------------------------------------------------------------
__global__ __launch_bounds__(128) void gemm_wmma(
    const float* __restrict__ A, int lda,
    const float* __restrict__ W, int ldw,
    const float* __restrict__ bias,
    float* __restrict__ C, int ldc,
    int M, int N, int K, int relu)
{
  const int lane = threadIdx.x;
  const int g = lane >> 4, ln = lane & 15;
  const int m0 = blockIdx.x * 32;
  const int wt = blockIdx.y * blockDim.y + threadIdx.y;   // wave column
  const int n0 = wt * 64;
  if (n0 >= N) return;           // wave-uniform
  (void)M;

  const float* arow0 = A + (size_t)(m0 + ln) * lda + 2 * g;
  const float* arow1 = A + (size_t)(m0 + 16 + ln) * lda + 2 * g;
  const float* wrow[4];
  bool nv[4];
  #pragma unroll
  for (int j = 0; j < 4; ++j) {
    const int nj = n0 + 16 * j;
    nv[j] = (nj < N);                       // wave-uniform
    wrow[j] = W + (size_t)((nv[j] ? nj : 0) + ln) * ldw + 2 * g;
  }

  v8f acc[2][4];
  #pragma unroll
  for (int i = 0; i < 2; ++i)
    #pragma unroll
    for (int j = 0; j < 4; ++j)
      acc[i][j] = (v8f){0.f,0.f,0.f,0.f,0.f,0.f,0.f,0.f};

  // Prologue: load fragments for k = 0.
  v2f a0 = *(const v2f*)(arow0);
  v2f a1 = *(const v2f*)(arow1);
  v2f bb[4];
  #pragma unroll
  for (int j = 0; j < 4; ++j) bb[j] = *(const v2f*)(wrow[j]);

  for (int k = 0; k < K; k += 4) {
    // Branch-free prefetch index: last iteration re-loads in-bounds data.
    const int kn = (k + 4 < K) ? (k + 4) : k;
    const v2f na0 = *(const v2f*)(arow0 + kn);
    const v2f na1 = *(const v2f*)(arow1 + kn);
    v2f nb[4];
    #pragma unroll
    for (int j = 0; j < 4; ++j) nb[j] = *(const v2f*)(wrow[j] + kn);

    // 8 WMMAs on the resident fragments.
    #pragma unroll
    for (int j = 0; j < 4; ++j) {
      acc[0][j] = wmma4(a0, bb[j], acc[0][j]);
      acc[1][j] = wmma4(a1, bb[j], acc[1][j]);
    }

    a0 = na0; a1 = na1;
    #pragma unroll
    for (int j = 0; j < 4; ++j) bb[j] = nb[j];
  }

  #pragma unroll
  for (int j = 0; j < 4; ++j) {
    if (!nv[j]) continue;                   // wave-uniform
    const int n = n0 + 16 * j + ln;
    const float bn = bias ? bias[n] : 0.f;
    #pragma unroll
    for (int i = 0; i < 2; ++i) {
      float* crow = C + (size_t)(m0 + 16 * i + g * 8) * ldc + n;
      #pragma unroll
      for (int v = 0; v < 8; ++v) {
        float r = acc[i][j][v] + bn;
        if (relu) r = fmaxf(r, 0.f);
        crow[(size_t)v * ldc] = r;
      }
    }
  }
}

// ---------------------------------------------------------------------------
// Flash attention over qkv buffer laid out [t = q*BS + b][768]
//   Q at col 0, K at col 256, V at col 512; head h occupies 32 cols h*32.
// One wave per (16-row q-tile, b, h). Online softmax, P routed via LDS.
// out[t][256] with feature h*32+d  (== o.transpose(2,0,1,3).reshape layout)
// ---------------------------------------------------------------------------
__global__ __launch_bounds__(32) void flash_attn(
    const float* __restrict__ qkv, float* __restrict__ out)
{
  __shared__ float sP[16][17];
  const int lane = threadIdx.x;
  const int g = lane >> 4, ln = lane & 15;
  const int qt = blockIdx.x;
  const int b  = blockIdx.y;
  const int h  = blockIdx.z;
  const float SCALE = 0.17677669529663687f;   // 1/sqrt(32)

  // Preload Q fragments (16x32 tile: 8 K-steps)
  const int qi_a = qt * 16 + ln;
  v2f aq[8];
  if (qi_a < NQ) {
    const float* qp = qkv + ((size_t)qi_a * BS + b) * 768 + h * 32 + 2 * g;
    #pragma unroll
    for (int s = 0; s < 8; ++s) aq[s] = *(const v2f*)(qp + 4 * s);
  } else {
    #pragma unroll
    for (int s = 0; s < 8; ++s) aq[s] = (v2f){0.f, 0.f};
  }

  v8f o0 = {0.f,0.f,0.f,0.f,0.f,0.f,0.f,0.f};
  v8f o1 = {0.f,0.f,0.f,0.f,0.f,0.f,0.f,0.f};
  float Mrow[8], Lrow[8];
  #pragma unroll
  for (int v = 0; v < 8; ++v) { Mrow[v] = -INFINITY; Lrow[v] = 0.f; }

  for (int kt = 0; kt < QT; ++kt) {
    const int kj = kt * 16 + ln;
    const bool cv = (kj < NQ);
    const float* kp = qkv + ((size_t)(cv ? kj : 0) * BS + b) * 768 + 256 + h * 32 + 2 * g;

    // S tile = Q * K^T
    v8f s = {0.f,0.f,0.f,0.f,0.f,0.f,0.f,0.f};
    #pragma unroll
    for (int t8 = 0; t8 < 8; ++t8) {
      v2f bk = cv ? *(const v2f*)(kp + 4 * t8) : (v2f){0.f, 0.f};
      s = wmma4(aq[t8], bk, s);
    }

    // Online softmax per row (rows v+8g live across the 16 lanes of half g)
    float p[8];
    #pragma unroll
    for (int v = 0; v < 8; ++v) {
      float sv = cv ? s[v] * SCALE : -INFINITY;
      float mx = half_max16(sv);
      float Mn = fmaxf(Mrow[v], mx);
      float pe = cv ? expf(sv - Mn) : 0.f;
      float su = half_sum16(pe);
      float f  = expf(Mrow[v] - Mn);   // 0 on first tile (exp(-inf))
      Lrow[v] = Lrow[v] * f + su;
      Mrow[v] = Mn;
      o0[v] *= f; o1[v] *= f;
      p[v] = pe;
    }

    // P tile -> LDS (C-layout -> row-major), then O += P * V via WMMA
    __syncthreads();
    #pragma unroll
    for (int v = 0; v < 8; ++v) sP[v + 8 * g][ln] = p[v];
    __syncthreads();

    #pragma unroll
    for (int s4 = 0; s4 < 4; ++s4) {
      const int kk = 4 * s4 + 2 * g;
      v2f ap = { sP[ln][kk], sP[ln][kk + 1] };
      const int kj0 = kt * 16 + kk, kj1 = kj0 + 1;
      const size_t r0 = ((size_t)(kj0 < NQ ? kj0 : 0) * BS + b) * 768 + 512 + h * 32;
      const size_t r1 = ((size_t)(kj1 < NQ ? kj1 : 0) * BS + b) * 768 + 512 + h * 32;
      const bool ok0 = (kj0 < NQ), ok1 = (kj1 < NQ);
      v2f b0 = { ok0 ? qkv[r0 + ln]      : 0.f, ok1 ? qkv[r1 + ln]      : 0.f };
      v2f b1 = { ok0 ? qkv[r0 + 16 + ln] : 0.f, ok1 ? qkv[r1 + 16 + ln] : 0.f };
      o0 = wmma4(ap, b0, o0);
      o1 = wmma4(ap, b1, o1);
    }
    __syncthreads();
  }

  // Normalize and store
  #pragma unroll
  for (int v = 0; v < 8; ++v) {
    const int qi = qt * 16 + v + 8 * g;
    if (qi < NQ) {
      const float inv = 1.0f / Lrow[v];
      float* op = out + ((size_t)qi * BS + b) * EE + h * 32;
      op[ln]      = o0[v] * inv;
      op[16 + ln] = o1[v] * inv;
    }
  }
}

// ---------------------------------------------------------------------------
// out = LayerNorm(a + r) * w + beta   over rows of length 256
// block (32,8): one wave per row
// ---------------------------------------------------------------------------
__global__ __launch_bounds__(256) void ln_add_kernel(
    const float* __restrict__ a, const float* __restrict__ r,
    const float* __restrict__ w, const float* __restrict__ bet,
    float* __restrict__ out, int rows)
{
  const int row = blockIdx.x * blockDim.y + threadIdx.y;
  if (row >= rows) return;
  const int lane = threadIdx.x;
  const float* pa = a + (size_t)row * EE;
  const float* pr = r + (size_t)row * EE;
  float x[8]; float s = 0.f;
  #pragma unroll
  for (int i = 0; i < 8; ++i) { x[i] = pa[lane + 32*i] + pr[lane + 32*i]; s += x[i]; }
  s = wave_sum32(s);
  const float mu = s * (1.f / EE);
  float vs = 0.f;
  #pragma unroll
  for (int i = 0; i < 8; ++i) { float d = x[i] - mu; vs += d * d; }
  vs = wave_sum32(vs);
  const float rstd = rsqrtf(vs * (1.f / EE) + 1e-5f);
  float* po = out + (size_t)row * EE;
  #pragma unroll
  for (int i = 0; i < 8; ++i)
    po[lane + 32*i] = (x[i] - mu) * rstd * w[lane + 32*i] + bet[lane + 32*i];
}

__global__ void add2_kernel(const float* __restrict__ a,
                            const float* __restrict__ b,
                            float* __restrict__ o, size_t n)
{
  size_t i = (size_t)blockIdx.x * blockDim.x + threadIdx.x;
  if (i < n) o[i] = a[i] + b[i];
}

// ---------------------------------------------------------------------------
// MSDA bilinear sampling.  pv[t2 = nv*BS + b][256] (feature h*32+d).
// block (32,8): lane=d, ty=h; grid (NQ, BS). Output [t=q*BS+b][256].
// ---------------------------------------------------------------------------
__device__ __forceinline__ float sample_pv(const float* __restrict__ pv,
                                           int b, int h, int d, int xi, int yi)
{
  if (xi < 0 || xi >= SW || yi < 0 || yi >= SH) return 0.f;
  return pv[(((size_t)yi * SW + xi) * BS + b) * EE + h * DH + d];
}

__global__ __launch_bounds__(256) void msda_kernel(
    const float* __restrict__ pv, const float* __restrict__ offb,
    const float* __restrict__ awb, const float* __restrict__ refp,
    float* __restrict__ out)
{
  const int d = threadIdx.x;
  const int h = threadIdx.y;
  const int q = blockIdx.x, b = blockIdx.y;
  const size_t t = (size_t)q * BS + b;
  const float rx = refp[((size_t)b * NQ + q) * 2 + 0];
  const float ry = refp[((size_t)b * NQ + q) * 2 + 1];

  float lg[4];
  #pragma unroll
  for (int p = 0; p < 4; ++p) lg[p] = awb[t * 32 + h * 4 + p];
  float mx = fmaxf(fmaxf(lg[0], lg[1]), fmaxf(lg[2], lg[3]));
  float se = 0.f;
  #pragma unroll
  for (int p = 0; p < 4; ++p) { lg[p] = expf(lg[p] - mx); se += lg[p]; }
  const float inv = 1.f / se;

  float acc = 0.f;
  #pragma unroll
  for (int p = 0; p < 4; ++p) {
    const float aw = lg[p] * inv;
    const float ox = offb[t * 64 + (h * 4 + p) * 2 + 0];
    const float oy = offb[t * 64 + (h * 4 + p) * 2 + 1];
    // loc = ref + off/[W,H];  x = loc.x*W - 0.5, y = loc.y*H - 0.5
    const float x = (rx + ox * (1.f / SW)) * SW - 0.5f;
    const float y = (ry + oy * (1.f / SH)) * SH - 0.5f;
    const float x0f = floorf(x), y0f = floorf(y);
    const float wx = x - x0f, wy = y - y0f;
    const int x0 = (int)x0f, y0 = (int)y0f;
    const float v00 = sample_pv(pv, b, h, d, x0,     y0    );
    const float v01 = sample_pv(pv, b, h, d, x0 + 1, y0    );
    const float v10 = sample_pv(pv, b, h, d, x0,     y0 + 1);
    const float v11 = sample_pv(pv, b, h, d, x0 + 1, y0 + 1);
    acc += aw * (v00 * (1.f - wx) * (1.f - wy) + v01 * wx * (1.f - wy) +
                 v10 * (1.f - wx) * wy         + v11 * wx * wy);
  }
  out[t * EE + h * DH + d] = acc;
}

// ---------------------------------------------------------------------------
extern "C" void kernel_launch(void* const* d_in, const int* in_sizes, int n_in,
                              void* d_out, int out_size, void* d_ws, size_t ws_size,
                              hipStream_t stream)
{
  (void)in_sizes; (void)n_in; (void)out_size; (void)ws_size;
  const float* query = (const float*)d_in[0];
  const float* qpos  = (const float*)d_in[1];
  const float* value = (const float*)d_in[2];
  const float* refp  = (const float*)d_in[3];
  const float* in_w  = (const float*)d_in[4];
  const float* in_b  = (const float*)d_in[5];
  const float* out_w = (const float*)d_in[6];
  const float* out_b = (const float*)d_in[7];
  const float* ln1w  = (const float*)d_in[8];
  const float* ln1b  = (const float*)d_in[9];
  const float* ln2w  = (const float*)d_in[10];
  const float* ln2b  = (const float*)d_in[11];
  const float* ln3w  = (const float*)d_in[12];
  const float* ln3b  = (const float*)d_in[13];
  const float* off_w = (const float*)d_in[14];
  const float* off_b = (const float*)d_in[15];
  const float* aw_w  = (const float*)d_in[16];
  const float* aw_b  = (const float*)d_in[17];
  const float* vp_w  = (const float*)d_in[18];
  const float* vp_b  = (const float*)d_in[19];
  const float* op_w  = (const float*)d_in[20];
  const float* op_b  = (const float*)d_in[21];
  const float* f1_w  = (const float*)d_in[22];
  const float* f1_b  = (const float*)d_in[23];
  const float* f2_w  = (const float*)d_in[24];
  const float* f2_b  = (const float*)d_in[25];
  float* outp = (float*)d_out;

  const size_t T = (size_t)NQ * BS;          // 8000 tokens
  float* ws   = (float*)d_ws;
  float* bufA = ws;                          // T*E floats (multi-use)
  float* bufB = bufA + T * EE;               // T*3E floats (multi-use)
  float* bufP = bufB + T * 3 * EE;           // (NV*BS)*E floats (pv / ffn_h)
  float* mha  = bufB;                        // [0, T*E)
  float* xbuf = bufB + T * EE;               // [T*E, 2T*E)
  float* offb = bufB + 2 * T * EE;           // T*64
  float* awb  = offb + T * 64;               // T*32

  const dim3 gblk(32, 4);
  const dim3 lnblk(32, 8);
  const dim3 msblk(32, 8);
  // grid helper for gemm_wmma: (M/32, ceil(ceil(N/64)/4))
  #define GEMM_GRID(M, N) dim3((unsigned)((M) / 32), (unsigned)((((N) + 63) / 64 + 3) / 4))

  // 1. qp = query + query_pos
  add2_kernel<<<dim3((unsigned)((T * EE + 255) / 256)), 256, 0, stream>>>(
      query, qpos, bufA, T * EE);
  // 2. qkv = qp @ in_proj_w^T + in_proj_b   [T x 768]
  gemm_wmma<<<GEMM_GRID(T, 768), gblk, 0, stream>>>(
      bufA, EE, in_w, EE, in_b, bufB, 768, (int)T, 768, EE, 0);
  // 3. flash attention -> attn_out (bufA)
  flash_attn<<<dim3(QT, BS, HH), 32, 0, stream>>>(bufB, bufA);
  // 4. mha = attn_out @ out_proj_w^T + b
  gemm_wmma<<<GEMM_GRID(T, EE), gblk, 0, stream>>>(
      bufA, EE, out_w, EE, out_b, mha, EE, (int)T, EE, EE, 0);
  // 5. x = LN1(query + mha)
  ln_add_kernel<<<dim3((unsigned)(T / 8)), lnblk, 0, stream>>>(
      query, mha, ln1w, ln1b, xbuf, (int)T);
  // 6. xq = x + query_pos
  add2_kernel<<<dim3((unsigned)((T * EE + 255) / 256)), 256, 0, stream>>>(
      xbuf, qpos, bufA, T * EE);
  // 7. off = xq @ off_w^T + off_b   [T x 64]
  gemm_wmma<<<GEMM_GRID(T, 64), gblk, 0, stream>>>(
      bufA, EE, off_w, EE, off_b, offb, 64, (int)T, 64, EE, 0);
  // 8. aw logits = xq @ aw_w^T + aw_b   [T x 32]
  gemm_wmma<<<GEMM_GRID(T, 32), gblk, 0, stream>>>(
      bufA, EE, aw_w, EE, aw_b, awb, 32, (int)T, 32, EE, 0);
  // 9. pv = value @ vproj_w^T + b    [(NV*BS) x 256]
  gemm_wmma<<<GEMM_GRID(NV * BS, EE), gblk, 0, stream>>>(
      value, EE, vp_w, EE, vp_b, bufP, EE, NV * BS, EE, EE, 0);
  // 10. MSDA bilinear sampling -> msda_out (bufA)
  msda_kernel<<<dim3(NQ, BS), msblk, 0, stream>>>(bufP, offb, awb, refp, bufA);
  // 11. ms = msda_out @ oproj_w^T + b -> mha region
  gemm_wmma<<<GEMM_GRID(T, EE), gblk, 0, stream>>>(
      bufA, EE, op_w, EE, op_b, mha, EE, (int)T, EE, EE, 0);
  // 12. x2 = LN2(x + ms) -> bufA
  ln_add_kernel<<<dim3((unsigned)(T / 8)), lnblk, 0, stream>>>(
      xbuf, mha, ln2w, ln2b, bufA, (int)T);
  // 13. ffn_h = relu(x2 @ ffn_w1^T + b1)  [T x 512] -> bufP
  gemm_wmma<<<GEMM_GRID(T, FFND), gblk, 0, stream>>>(
      bufA, EE, f1_w, EE, f1_b, bufP, FFND, (int)T, FFND, EE, 1);
  // 14. ffn_o = ffn_h @ ffn_w2^T + b2 -> mha region
  gemm_wmma<<<GEMM_GRID(T, EE), gblk, 0, stream>>>(
      bufP, FFND, f2_w, FFND, f2_b, mha, EE, (int)T, EE, FFND, 0);
  // 15. out = LN3(x2 + ffn_o)
  ln_add_kernel<<<dim3((unsigned)(T / 8)), lnblk, 0, stream>>>(
      bufA, mha, ln3w, ln3b, outp, (int)T);
  #undef GEMM_GRID
}